// DistanceLayer_7550552507241
// MI455X (gfx1250) — compile-verified
//
#include <hip/hip_runtime.h>
#include <hip/hip_bf16.h>
#include <math.h>

typedef __attribute__((ext_vector_type(2))) float v2f;
typedef __attribute__((ext_vector_type(4))) float v4f;
typedef __attribute__((ext_vector_type(8))) float v8f;

#define B_   8
#define N_   2048
#define IND  256
#define MD   128
#define NT_  (N_ / 16)            // 128 j-tiles

// --------------------------------------------------------------------------
// gfx1250 async global->LDS copy (tracked by ASYNCcnt, data lands in LDS
// without touching VGPRs).  lds_off = low 32 bits of the generic pointer to
// the __shared__ destination (== workgroup LDS byte offset on amdgcn).
// --------------------------------------------------------------------------
__device__ __forceinline__ void async_ld_b128(uint32_t lds_off, const float* g) {
    asm volatile("global_load_async_to_lds_b128 %0, %1, off"
                 :: "v"(lds_off), "v"((uint64_t)(uintptr_t)g)
                 : "memory");
}

// ---------------------------------------------------------------------------
// Kernel 1: projection GEMM.  out[row,n] = sum_d x[row,d]*W[d,n] + bias[n].
// One wave owns a 16-row tile and computes ALL 128 columns (8 v8f accums),
// so x (16 MB) is read exactly once from HBM; W (128 KB) stays L2/WGP$-hot.
// blockIdx.z: 0 -> Q, 1 -> K.
// ---------------------------------------------------------------------------
__global__ __launch_bounds__(256) void proj_kernel(
    const float* __restrict__ x,
    const float* __restrict__ Wq, const float* __restrict__ bq,
    const float* __restrict__ Wk, const float* __restrict__ bk,
    float* __restrict__ Qs, float* __restrict__ Ks)
{
    const int lane = threadIdx.x & 31;
    const int wave = threadIdx.x >> 5;
    const int half = lane >> 4;           // A/B frags: lanes 16-31 hold K=2,3
    const int l16  = lane & 15;

    const float* W;  const float* bias;  float* out;
    if (blockIdx.z == 0) { W = Wq; bias = bq; out = Qs; }
    else                 { W = Wk; bias = bk; out = Ks; }

    const int tm = blockIdx.x * 8 + wave;          // 1024 row tiles
    const int m0 = tm * 16;
    const float* xrow = x + (size_t)(m0 + l16) * IND;

    v8f c[8] = {};                                 // 8 column tiles
    for (int kb = 0; kb < IND; kb += 4) {
        const int ka = kb + (half << 1);
        v2f a = *(const v2f*)(xrow + ka);          // one b64 x-load / step
        const float* w0 = W + (size_t)ka * MD + l16;
        const float* w1 = w0 + MD;
        #pragma unroll
        for (int tn = 0; tn < 8; ++tn) {           // reuse A across 8 WMMAs
            v2f bf;
            bf.x = w0[tn * 16];
            bf.y = w1[tn * 16];
            c[tn] = __builtin_amdgcn_wmma_f32_16x16x4_f32(
                        false, a, false, bf, (short)0, c[tn], false, false);
        }
    }

    #pragma unroll
    for (int tn = 0; tn < 8; ++tn) {
        const int   ncol = tn * 16 + l16;
        const float bv   = bias[ncol];
        #pragma unroll
        for (int r = 0; r < 8; ++r) {
            const int row = m0 + r + half * 8;     // C layout: VGPR r
            out[(size_t)row * MD + ncol] = c[tn][r] + bv;
        }
    }
}

// ---------------------------------------------------------------------------
// Kernel 2: fused scores * gaussian prior + online softmax + dist-weighted sum.
// Block = 128 threads (4 waves); each wave owns one 16-row i-tile, all waves
// share K tiles double-buffered through LDS via async global->LDS loads.
// Per-lane online softmax (no shuffles in the hot loop); 16-lane merge at end.
// ---------------------------------------------------------------------------
__global__ __launch_bounds__(128) void attn_kernel(
    const float* __restrict__ Qs, const float* __restrict__ Ks,
    const float* __restrict__ prior_mean, const float* __restrict__ prior_std,
    float* __restrict__ out)
{
    const int lane = threadIdx.x & 31;
    const int wave = threadIdx.x >> 5;    // 0..3
    const int half = lane >> 4;
    const int l16  = lane & 15;

    const int b  = blockIdx.y;
    const int i0 = blockIdx.x * 64 + wave * 16;    // this wave's i-tile

    // 132-float row pitch -> ds_load_b64 B-frag reads are bank-conflict-free
    __shared__ float kt[2][16][132];

    const float mean  = prior_mean[0];
    const float istd  = 1.0f / prior_std[0];
    const float coef  = 0.3989422804014327f * istd;   // 1/(std*sqrt(2*pi))
    const float scale = 0.08838834764831845f;         // 1/sqrt(128)

    const float* Kb = Ks + (size_t)b * N_ * MD;

    // Preload this wave's 16x128 Q tile as 32 A-fragments (64 VGPRs/lane).
    v2f afrag[32];
    {
        const float* qrow = Qs + ((size_t)b * N_ + i0 + l16) * MD;
        #pragma unroll
        for (int kk = 0; kk < 32; ++kk)
            afrag[kk] = *(const v2f*)(qrow + kk * 4 + (half << 1));
    }

    // Per-lane online softmax state for rows r (lanes 0-15) / r+8 (16-31).
    float m[8], l[8], acc[8];
    #pragma unroll
    for (int r = 0; r < 8; ++r) { m[r] = -3.0e38f; l[r] = 0.0f; acc[r] = 0.0f; }

    // --- stage tile 0 : 128 threads x 4 async b128 chunks = 16x128 floats ---
    {
        const float* src = Kb;
        #pragma unroll
        for (int p = 0; p < 4; ++p) {
            const int cidx = p * 128 + threadIdx.x;    // float4 chunk id
            const int row  = cidx >> 5;
            const int col  = (cidx & 31) << 2;
            async_ld_b128((uint32_t)(uintptr_t)&kt[0][row][col],
                          src + (size_t)cidx * 4);
        }
    }

    for (int t = 0; t < NT_; ++t) {
        const int buf = t & 1;

        if (t + 1 < NT_) {               // prefetch next tile into other buffer
            const float* src = Kb + (size_t)(t + 1) * (16 * MD);
            #pragma unroll
            for (int p = 0; p < 4; ++p) {
                const int cidx = p * 128 + threadIdx.x;
                const int row  = cidx >> 5;
                const int col  = (cidx & 31) << 2;
                async_ld_b128((uint32_t)(uintptr_t)&kt[buf ^ 1][row][col],
                              src + (size_t)cidx * 4);
            }
            asm volatile("s_wait_asynccnt 0x4" ::: "memory"); // tile t landed
        } else {
            asm volatile("s_wait_asynccnt 0x0" ::: "memory");
        }
        __syncthreads();                 // tile t visible to all 4 waves

        const float* kls = &kt[buf][l16][0];           // lane = j column
        v8f c = {};
        #pragma unroll
        for (int kk = 0; kk < 32; ++kk) {
            const int ka = kk * 4 + (half << 1);
            v2f bf = *(const v2f*)(kls + ka);          // conflict-free ds b64
            c = __builtin_amdgcn_wmma_f32_16x16x4_f32(
                    false, afrag[kk], false, bf, (short)0, c, false, false);
        }

        const int j = t * 16 + l16;
        #pragma unroll
        for (int r = 0; r < 8; ++r) {
            const int   i  = i0 + r + half * 8;
            const float d  = (float)(j - i);
            const float z  = (d - mean) * istd;
            const float g  = coef * __expf(-0.5f * z * z);   // gaussian prior
            const float s  = c[r] * scale * g;               // score
            // one-exp online update: one of the two rescales is exp(0)
            const float dl   = s - m[r];
            const float ex   = __expf(-fabsf(dl));
            const bool  pos  = dl > 0.0f;
            const float corr = pos ? ex : 1.0f;
            const float e    = pos ? 1.0f : ex;
            l[r]   = l[r]   * corr + e;
            acc[r] = acc[r] * corr + e * d;
            m[r]   = fmaxf(m[r], s);
        }
        __syncthreads();                 // all waves done before buf reuse
    }

    // Merge the 16 per-lane partial states of each row (once per kernel).
    #pragma unroll
    for (int r = 0; r < 8; ++r) {
        float mm = m[r], ll = l[r], aa = acc[r];
        #pragma unroll
        for (int mask = 1; mask <= 8; mask <<= 1) {   // stays inside 16-lane half
            const float om = __shfl_xor(mm, mask);
            const float ol = __shfl_xor(ll, mask);
            const float oa = __shfl_xor(aa, mask);
            const float mn = fmaxf(mm, om);
            const float c0 = __expf(mm - mn);
            const float c1 = __expf(om - mn);
            ll = ll * c0 + ol * c1;
            aa = aa * c0 + oa * c1;
            mm = mn;
        }
        if (l16 == 0)
            out[(size_t)b * N_ + i0 + r + half * 8] = aa / ll;
    }
}

// ---------------------------------------------------------------------------
extern "C" void kernel_launch(void* const* d_in, const int* in_sizes, int n_in,
                              void* d_out, int out_size, void* d_ws, size_t ws_size,
                              hipStream_t stream) {
    const float* x  = (const float*)d_in[0];
    const float* Wq = (const float*)d_in[1];
    const float* bq = (const float*)d_in[2];
    const float* Wk = (const float*)d_in[3];
    const float* bk = (const float*)d_in[4];
    const float* pm = (const float*)d_in[5];
    const float* ps = (const float*)d_in[6];
    float* out = (float*)d_out;

    float* Qs = (float*)d_ws;                       // 8 MB
    float* Ks = Qs + (size_t)B_ * N_ * MD;          // 8 MB

    dim3 g1(128, 1, 2);                             // 8 waves/block x {Q,K}
    proj_kernel<<<g1, 256, 0, stream>>>(x, Wq, bq, Wk, bk, Qs, Ks);

    dim3 g2(N_ / 64, B_);                           // 4 i-tiles per block
    attn_kernel<<<g2, 128, 0, stream>>>(Qs, Ks, pm, ps, out);
}